// MultiheadSelfAttention2D_69904887710022
// MI455X (gfx1250) — compile-verified
//
#include <hip/hip_runtime.h>

#define C_IN  256
#define AC    64
#define HEADS 16
#define HD    4
#define L     4096   // 64*64
#define JCH   2048   // LDS chunk length along l

#define LOG2E 1.44269504088896340736f

typedef __attribute__((ext_vector_type(2))) float v2f;
typedef __attribute__((ext_vector_type(8))) float v8f;
typedef __attribute__((ext_vector_type(4))) int   v4i;

// workspace layout (floats)
#define WS_Q 0
#define WS_K (AC * L)
#define WS_V (2 * AC * L)
#define WS_Z (3 * AC * L)
#define WS_O (3 * AC * L + HEADS * L)

// ---------------------------------------------------------------------------
// Raw hardware 2^x (v_exp_f32)
// ---------------------------------------------------------------------------
__device__ __forceinline__ float fast_exp2(float x) {
#if __has_builtin(__builtin_amdgcn_exp2f)
    return __builtin_amdgcn_exp2f(x);
#else
    return exp2f(x);
#endif
}

// ---------------------------------------------------------------------------
// CDNA5 async global->LDS copy (ASYNCcnt-tracked, no VGPR staging)
// ---------------------------------------------------------------------------
__device__ __forceinline__ void async_copy_b128(void* lds_dst, const void* gsrc) {
#if __has_builtin(__builtin_amdgcn_global_load_async_to_lds_b128)
    typedef __attribute__((address_space(1))) v4i* gptr_t;   // global (AS1)
    typedef __attribute__((address_space(3))) v4i* lptr_t;   // LDS (AS3)
    __builtin_amdgcn_global_load_async_to_lds_b128(
        (gptr_t)(uintptr_t)gsrc,
        (lptr_t)(uintptr_t)lds_dst,
        0, 0);
#else
    unsigned lo = (unsigned)(uintptr_t)lds_dst;
    unsigned long long ga = (unsigned long long)(uintptr_t)gsrc;
    asm volatile("global_load_async_to_lds_b128 %0, %1, off"
                 :: "v"(lo), "v"(ga) : "memory");
#endif
}

__device__ __forceinline__ void wait_async0() {
#if __has_builtin(__builtin_amdgcn_s_wait_asynccnt)
    __builtin_amdgcn_s_wait_asynccnt(0);
#else
    asm volatile("s_wait_asynccnt 0x0" ::: "memory");
#endif
}

// ---------------------------------------------------------------------------
// Kernel 1: QKV projections (1x1 conv == channel GEMM) + bias + L2-norm(Q,K)
// Q is additionally scaled by log2(e): scores then feed v_exp_f32 (2^x)
// directly with no per-element multiply in the hot loops.
// ---------------------------------------------------------------------------
__global__ void qkv_kernel(const float* __restrict__ X,
                           const float* __restrict__ Wq, const float* __restrict__ bq,
                           const float* __restrict__ Wk, const float* __restrict__ bk,
                           const float* __restrict__ Wv, const float* __restrict__ bv,
                           float* __restrict__ Qg, float* __restrict__ Kg,
                           float* __restrict__ Vg) {
    __shared__ float Xs[C_IN * 16];   // X[:, lbase..lbase+15]
    __shared__ float Ys[192 * 16];    // raw projections before norm

    const int t = threadIdx.x;
    const int lbase = blockIdx.x * 16;

    {
        const int i = t & 15;
        const int c0 = t >> 4;
        #pragma unroll
        for (int s = 0; s < 16; ++s) {
            const int c = c0 + s * 16;
            Xs[c * 16 + i] = X[c * L + lbase + i];
        }
    }
    __syncthreads();

    if (t < 192) {
        const int mat = t >> 6;      // 0=Q 1=K 2=V
        const int o   = t & 63;
        const float* Wm = (mat == 0 ? Wq : (mat == 1 ? Wk : Wv)) + o * C_IN;
        const float bias = (mat == 0 ? bq : (mat == 1 ? bk : bv))[o];
        float acc[16];
        #pragma unroll
        for (int i = 0; i < 16; ++i) acc[i] = bias;
        const float4* w4 = reinterpret_cast<const float4*>(Wm);
        for (int c = 0; c < C_IN; c += 4) {
            const float4 w = w4[c >> 2];
            #pragma unroll
            for (int i = 0; i < 16; ++i) {
                acc[i] = fmaf(w.x, Xs[(c + 0) * 16 + i], acc[i]);
                acc[i] = fmaf(w.y, Xs[(c + 1) * 16 + i], acc[i]);
                acc[i] = fmaf(w.z, Xs[(c + 2) * 16 + i], acc[i]);
                acc[i] = fmaf(w.w, Xs[(c + 3) * 16 + i], acc[i]);
            }
        }
        #pragma unroll
        for (int i = 0; i < 16; ++i) Ys[t * 16 + i] = acc[i];
    }
    __syncthreads();

    if (t < 192) {
        const int mat = t >> 6;
        const int o   = t & 63;          // o = head*4 + d (matches reshape)
        const int d   = o & 3;
        float* dst = (mat == 0 ? Qg : (mat == 1 ? Kg : Vg));
        #pragma unroll
        for (int i = 0; i < 16; ++i) {
            float y = Ys[t * 16 + i];
            if (mat < 2) {
                const int base = (t - d) * 16 + i;
                float s = 0.f;
                #pragma unroll
                for (int dd = 0; dd < 4; ++dd) {
                    const float q = Ys[base + dd * 16];
                    s = fmaf(q, q, s);
                }
                y = y / fmaxf(sqrtf(s), 1e-12f);
                if (mat == 0) y *= LOG2E;   // fold exp->exp2 conversion into Q
            }
            dst[o * L + lbase + i] = y;
        }
    }
}

// ---------------------------------------------------------------------------
// Kernel 2: softmax denominators Z[h][j] = sum_m exp(q_j . k_m)
//           (computed as sum_m 2^(q'_j . k_m) with q' = q * log2e)
// ---------------------------------------------------------------------------
__global__ void zrow_kernel(const float* __restrict__ Qg,
                            const float* __restrict__ Kg,
                            float* __restrict__ Zg) {
    extern __shared__ float Ks[];            // [HD][JCH]
    const int t  = threadIdx.x;
    const int h  = blockIdx.x;
    const int h4 = h * HD;
    const int jbase = blockIdx.y * 128 + (t >> 5) * 16;
    const int ln   = t & 31;
    const int mrow = ln & 15;
    const int d0   = (ln < 16) ? 0 : 2;

    // A fragment (Q rows jbase..jbase+15), loop-invariant over m
    v2f a;
    a.x = Qg[(h4 + d0 + 0) * L + jbase + mrow];
    a.y = Qg[(h4 + d0 + 1) * L + jbase + mrow];

    float zacc[8];
    #pragma unroll
    for (int r = 0; r < 8; ++r) zacc[r] = 0.f;

    for (int mc = 0; mc < L; mc += JCH) {
        __syncthreads();
        // async-stage K[d][mc..mc+JCH) -> Ks[d][...] (16B granules, coalesced)
        for (int idx = t; idx < HD * (JCH / 4); idx += 256) {
            const int d  = idx >> 9;          // / (JCH/4)
            const int m4 = idx & (JCH / 4 - 1);
            async_copy_b128(Ks + d * JCH + (m4 << 2),
                            Kg + (h4 + d) * L + mc + (m4 << 2));
        }
        wait_async0();
        __syncthreads();

        for (int mt = 0; mt < JCH; mt += 16) {
            v2f b;
            b.x = Ks[(d0 + 0) * JCH + mt + mrow];
            b.y = Ks[(d0 + 1) * JCH + mt + mrow];
            v8f c = {};
            c = __builtin_amdgcn_wmma_f32_16x16x4_f32(
                    false, a, false, b, (short)0, c, false, false);
            #pragma unroll
            for (int r = 0; r < 8; ++r) zacc[r] += fast_exp2(c[r]);
        }
    }

    // reduce across the 16 lanes of each half-wave (columns of the tile)
    #pragma unroll
    for (int r = 0; r < 8; ++r) {
        float v = zacc[r];
        v += __shfl_xor(v, 1);
        v += __shfl_xor(v, 2);
        v += __shfl_xor(v, 4);
        v += __shfl_xor(v, 8);
        zacc[r] = v;
    }
    if (ln == 0 || ln == 16) {
        const int off = (ln == 0) ? 0 : 8;
        #pragma unroll
        for (int r = 0; r < 8; ++r) Zg[h * L + jbase + off + r] = zacc[r];
    }
}

// ---------------------------------------------------------------------------
// Kernel 3: out[d,i] = sum_j (V[d,j]/Z_j) * 2^(q'_j . k_i)
// Q chunk async-staged as Qt[d][JCH]; V/Z staged manually (fused scale).
// ---------------------------------------------------------------------------
__global__ void attn_kernel(const float* __restrict__ Qg,
                            const float* __restrict__ Kg,
                            const float* __restrict__ Vg,
                            const float* __restrict__ Zg,
                            float* __restrict__ Og) {
    extern __shared__ float smem[];
    float* Qt = smem;                 // [HD][JCH]
    float* Vs = smem + HD * JCH;      // [JCH][4]  V/Z (transposed for float4 read)

    const int t  = threadIdx.x;
    const int h  = blockIdx.x;
    const int h4 = h * HD;
    const int w  = t >> 5;
    const int ln = t & 31;
    const int ncol = ln & 15;
    const int d0   = (ln < 16) ? 0 : 2;

    float acc[4][4];
    #pragma unroll
    for (int p = 0; p < 4; ++p)
        #pragma unroll
        for (int d = 0; d < 4; ++d) acc[p][d] = 0.f;

    for (int jc = 0; jc < L; jc += JCH) {
        __syncthreads();
        // async-stage Q rows for this chunk
        for (int idx = t; idx < HD * (JCH / 4); idx += 256) {
            const int d  = idx >> 9;
            const int j4 = idx & (JCH / 4 - 1);
            async_copy_b128(Qt + d * JCH + (j4 << 2),
                            Qg + (h4 + d) * L + jc + (j4 << 2));
        }
        // manual stage of V' = V / Z (needs the multiply)
        for (int idx = t; idx < HD * JCH; idx += 256) {
            const int d = idx >> 11;
            const int j = idx & (JCH - 1);
            const float rz = 1.0f / Zg[h * L + jc + j];
            Vs[j * 4 + d] = Vg[(h4 + d) * L + jc + j] * rz;
        }
        wait_async0();
        __syncthreads();

        for (int pass = 0; pass < 4; ++pass) {
            const int ibase = blockIdx.y * 512 + (pass * 8 + w) * 16;
            v2f b;                                   // K columns i (fixed per pass)
            b.x = Kg[(h4 + d0 + 0) * L + ibase + ncol];
            b.y = Kg[(h4 + d0 + 1) * L + ibase + ncol];

            for (int jt = 0; jt < JCH; jt += 16) {
                v2f a;                               // Q rows j (from LDS)
                a.x = Qt[(d0 + 0) * JCH + jt + ncol];
                a.y = Qt[(d0 + 1) * JCH + jt + ncol];
                v8f c = {};
                c = __builtin_amdgcn_wmma_f32_16x16x4_f32(
                        false, a, false, b, (short)0, c, false, false);
                const int rbase = jt + ((ln < 16) ? 0 : 8);
                #pragma unroll
                for (int r = 0; r < 8; ++r) {
                    const float p = fast_exp2(c[r]);
                    const float4 vv =
                        *reinterpret_cast<const float4*>(Vs + (rbase + r) * 4);
                    acc[pass][0] = fmaf(vv.x, p, acc[pass][0]);
                    acc[pass][1] = fmaf(vv.y, p, acc[pass][1]);
                    acc[pass][2] = fmaf(vv.z, p, acc[pass][2]);
                    acc[pass][3] = fmaf(vv.w, p, acc[pass][3]);
                }
            }
        }
    }

    // combine the two lane-halves (rows 0-7 vs 8-15 of every tile) and store
    #pragma unroll
    for (int pass = 0; pass < 4; ++pass) {
        const int i = blockIdx.y * 512 + (pass * 8 + w) * 16 + ncol;
        #pragma unroll
        for (int d = 0; d < 4; ++d) {
            const float v = acc[pass][d] + __shfl_xor(acc[pass][d], 16);
            if (ln < 16) Og[(h4 + d) * L + i] = v;
        }
    }
}

// ---------------------------------------------------------------------------
// Kernel 4: output projection + bias + residual
// ---------------------------------------------------------------------------
__global__ void outproj_kernel(const float* __restrict__ Og,
                               const float* __restrict__ Wo,
                               const float* __restrict__ bo,
                               const float* __restrict__ X,
                               float* __restrict__ Out) {
    __shared__ float As[AC * 16];
    const int t = threadIdx.x;
    const int lbase = blockIdx.x * 16;

    for (int idx = t; idx < AC * 16; idx += 256) {
        const int o = idx >> 4;
        const int i = idx & 15;
        As[idx] = Og[o * L + lbase + i];
    }
    __syncthreads();

    const int c = t;
    const float bias = bo[c];
    float acc[16];
    const float4* x4 = reinterpret_cast<const float4*>(X + c * L + lbase);
    #pragma unroll
    for (int i4 = 0; i4 < 4; ++i4) {
        const float4 xv = x4[i4];
        acc[i4 * 4 + 0] = bias + xv.x;
        acc[i4 * 4 + 1] = bias + xv.y;
        acc[i4 * 4 + 2] = bias + xv.z;
        acc[i4 * 4 + 3] = bias + xv.w;
    }
    const float4* w4 = reinterpret_cast<const float4*>(Wo + c * AC);
    for (int o = 0; o < AC; o += 4) {
        const float4 w = w4[o >> 2];
        #pragma unroll
        for (int i = 0; i < 16; ++i) {
            acc[i] = fmaf(w.x, As[(o + 0) * 16 + i], acc[i]);
            acc[i] = fmaf(w.y, As[(o + 1) * 16 + i], acc[i]);
            acc[i] = fmaf(w.z, As[(o + 2) * 16 + i], acc[i]);
            acc[i] = fmaf(w.w, As[(o + 3) * 16 + i], acc[i]);
        }
    }
    #pragma unroll
    for (int i = 0; i < 16; ++i) Out[c * L + lbase + i] = acc[i];
}

// ---------------------------------------------------------------------------
extern "C" void kernel_launch(void* const* d_in, const int* in_sizes, int n_in,
                              void* d_out, int out_size, void* d_ws, size_t ws_size,
                              hipStream_t stream) {
    (void)in_sizes; (void)n_in; (void)out_size; (void)ws_size;
    const float* X  = (const float*)d_in[0];
    const float* Wq = (const float*)d_in[1];
    const float* bq = (const float*)d_in[2];
    const float* Wk = (const float*)d_in[3];
    const float* bk = (const float*)d_in[4];
    const float* Wv = (const float*)d_in[5];
    const float* bv = (const float*)d_in[6];
    const float* Wo = (const float*)d_in[7];
    const float* bo = (const float*)d_in[8];
    float* out = (float*)d_out;

    float* ws = (float*)d_ws;
    float* Qg = ws + WS_Q;
    float* Kg = ws + WS_K;
    float* Vg = ws + WS_V;
    float* Zg = ws + WS_Z;
    float* Og = ws + WS_O;

    qkv_kernel<<<L / 16, 256, 0, stream>>>(X, Wq, bq, Wk, bk, Wv, bv, Qg, Kg, Vg);

    zrow_kernel<<<dim3(HEADS, L / 128), 256,
                  HD * JCH * sizeof(float), stream>>>(Qg, Kg, Zg);

    attn_kernel<<<dim3(HEADS, L / 512), 256,
                  2 * HD * JCH * sizeof(float), stream>>>(Qg, Kg, Vg, Zg, Og);

    outproj_kernel<<<L / 16, 256, 0, stream>>>(Og, Wo, bo, X, out);
}